// SelfAdaptiveFeatureExtractor_DV_48284022342030
// MI455X (gfx1250) — compile-verified
//
#include <hip/hip_runtime.h>
#include <hip/hip_bf16.h>

// ---------------- problem constants (from reference) ----------------
#define NPTS   500000
#define NTILES (NPTS / 16)          // 31250, exact
#define NZv    1
#define NYv    400
#define NXv    352
#define NB     4
#define CANVAS (NB * NZv * NYv * NXv)   // 563200
#define C0v    64
#define C1v    128
#define KPAD   16                   // 13 -> padded to 16
#define EPSV   0.001f
#define VXv    0.2f
#define VYv    0.2f
#define VZv    4.0f
#define XOFFv  0.1f
#define YOFFv  (-39.9f)
#define ZOFFv  (-1.0f)
#define INV_VOXVOL (1.0f / (VXv * VYv * VZv))

// register-blocking factors for the big GEMMs
#define RT     4                    // row tiles (of 16 rows) per wave
#define GW     2                    // waves per block in gemm1/gemm2
#define NWT    ((NTILES + RT - 1) / RT)    // wave-groups

typedef float v2f __attribute__((ext_vector_type(2)));
typedef float v8f __attribute__((ext_vector_type(8)));

// ---------------- helpers ----------------
__device__ __forceinline__ int voxel_of(const int* __restrict__ coors, int r) {
    int4 c = ((const int4*)coors)[r];
    return ((c.x * NZv + c.y) * NYv + c.z) * NXv + c.w;
}

// ---------------- setup / scatter kernels ----------------
__global__ void k_zero(float* __restrict__ p, size_t n) {
    size_t i = (size_t)blockIdx.x * blockDim.x + threadIdx.x;
    size_t s = (size_t)gridDim.x * blockDim.x;
    for (; i < n; i += s) p[i] = 0.0f;
}

__global__ void k_pad_w0(const float* __restrict__ W0, float* __restrict__ W0p) {
    int i = blockIdx.x * blockDim.x + threadIdx.x;           // 16*64
    if (i < KPAD * C0v) W0p[i] = (i < 13 * C0v) ? W0[i] : 0.0f;
}

__global__ void k_scatter_stats(const float* __restrict__ feat, const int* __restrict__ coors,
                                float* __restrict__ cnt, float* __restrict__ sx,
                                float* __restrict__ sy, float* __restrict__ sz, int n) {
    int r = blockIdx.x * blockDim.x + threadIdx.x;
    if (r >= n) return;
    float4 f = ((const float4*)feat)[r];
    int v = voxel_of(coors, r);
    atomicAdd(&cnt[v], 1.0f);
    atomicAdd(&sx[v], f.x);
    atomicAdd(&sy[v], f.y);
    atomicAdd(&sz[v], f.z);
}

__global__ void k_scatter_pdist(const float* __restrict__ feat, const int* __restrict__ coors,
                                const float* __restrict__ cnt, const float* __restrict__ sx,
                                const float* __restrict__ sy, const float* __restrict__ sz,
                                float* __restrict__ pd, int n) {
    int r = blockIdx.x * blockDim.x + threadIdx.x;
    if (r >= n) return;
    float4 f = ((const float4*)feat)[r];
    int v = voxel_of(coors, r);
    float safe = fmaxf(cnt[v], 1.0f);
    float dx = f.x - sx[v] / safe, dy = f.y - sy[v] / safe, dz = f.z - sz[v] / safe;
    atomicAdd(&pd[v], sqrtf(dx * dx + dy * dy + dz * dz));
}

__global__ void k_assemble(const float* __restrict__ feat, const int* __restrict__ coors,
                           const float* __restrict__ cnt, const float* __restrict__ sx,
                           const float* __restrict__ sy, const float* __restrict__ sz,
                           const float* __restrict__ pd, float* __restrict__ xpad,
                           float* __restrict__ fsp, int n) {
    int r = blockIdx.x * blockDim.x + threadIdx.x;
    if (r >= n) return;
    float4 f = ((const float4*)feat)[r];
    int4 c = ((const int4*)coors)[r];
    int v = ((c.x * NZv + c.y) * NYv + c.z) * NXv + c.w;
    float safe = fmaxf(cnt[v], 1.0f);
    float mx = sx[v] / safe, my = sy[v] / safe, mz = sz[v] / safe;
    float density = safe * INV_VOXVOL;
    float mdist = pd[v] / safe;
    float fc0 = f.x - mx, fc1 = f.y - my, fc2 = f.z - mz;
    float ce0 = f.x - ((float)c.w * VXv + XOFFv);
    float ce1 = f.y - ((float)c.z * VYv + YOFFv);
    float ce2 = f.z - ((float)c.y * VZv + ZOFFv);
    float dist = sqrtf(f.x * f.x + f.y * f.y + f.z * f.z);
    float4* xp = (float4*)(xpad + (size_t)r * KPAD);
    xp[0] = make_float4(f.x, f.y, f.z, f.w);
    xp[1] = make_float4(density, mdist, fc0, fc1);
    xp[2] = make_float4(fc2, ce0, ce1, ce2);
    xp[3] = make_float4(dist, 0.0f, 0.0f, 0.0f);
    fsp[2 * r] = density;
    fsp[2 * r + 1] = mdist;
}

// ---------------- BN stats / finalize ----------------
__global__ void k_bnstats(const float* __restrict__ Y, float* __restrict__ sum,
                          float* __restrict__ sq, int C, int n) {
    int t = blockIdx.x * blockDim.x + threadIdx.x;
    int total = gridDim.x * blockDim.x;     // multiple of C by construction
    int c = t % C;
    int stride = total / C;
    float ls = 0.0f, lq = 0.0f;
    for (int r = t / C; r < n; r += stride) {
        float v = Y[(size_t)r * C + c];
        ls += v;
        lq += v * v;
    }
    atomicAdd(&sum[c], ls);
    atomicAdd(&sq[c], lq);
}

__global__ void k_bnfinal(const float* __restrict__ sum, const float* __restrict__ sq,
                          const float* __restrict__ g, const float* __restrict__ b,
                          float* __restrict__ scale, float* __restrict__ shift,
                          int C, float invn) {
    int c = blockIdx.x * blockDim.x + threadIdx.x;
    if (c >= C) return;
    float m = sum[c] * invn;
    float var = sq[c] * invn - m * m;
    float sc = g[c] * rsqrtf(var + EPSV);
    scale[c] = sc;
    shift[c] = b[c] - m * sc;
}

// ---------------- segment-max scatter (post-BN ReLU => values >= 0) ----------------
__global__ void k_vmax(const float* __restrict__ y0, const float* __restrict__ scale0,
                       const float* __restrict__ shift0, const int* __restrict__ coors,
                       float* __restrict__ vmax, int n) {
    int idx = blockIdx.x * blockDim.x + threadIdx.x;
    if (idx >= n * C0v) return;
    int r = idx >> 6, c = idx & 63;
    float p = fmaxf(y0[idx] * scale0[c] + shift0[c], 0.0f);
    int v = voxel_of(coors, r);
    atomicMax((unsigned int*)&vmax[(size_t)v * C0v + c], __float_as_uint(p));
}

// ---------------- GEMM 0: xpad(N x 16) @ W0pad(16 x 64) -> y0 ----------------
// 4 waves/block, 4 row-tiles/wave; all B fragments (16x64) register-resident.
__global__ __launch_bounds__(128) void k_gemm0(const float* __restrict__ xpad,
                                               const float* __restrict__ Bw,
                                               float* __restrict__ y0, int ntiles) {
    __shared__ float smem[4 * RT * 16 * KPAD];               // 16 KB
    const int lane = threadIdx.x & 31, wv = threadIdx.x >> 5;
    const int base = (blockIdx.x * 4 + wv) * RT;             // first row tile of this wave
    float* As = smem + wv * (RT * 16 * KPAD);
    const int hl = lane >> 4, mn = lane & 15, kb = hl * 2;

    // stage A panels (guarded per tile; no early return before barrier)
#pragma unroll
    for (int rt = 0; rt < RT; ++rt) {
        if (base + rt < ntiles) {
            for (int idx = lane; idx < 16 * (KPAD / 4); idx += 32) {
                int row = idx / (KPAD / 4), c4 = idx % (KPAD / 4);
                ((float4*)As)[rt * (16 * KPAD / 4) + row * (KPAD / 4) + c4] =
                    ((const float4*)xpad)[(size_t)((base + rt) * 16 + row) * (KPAD / 4) + c4];
            }
        }
    }
    __syncthreads();
    if (base >= ntiles) return;

    // hoist all B fragments: 4 col-tiles x 4 k-steps
    v2f bf[4][KPAD / 4];
#pragma unroll
    for (int ct = 0; ct < 4; ++ct)
#pragma unroll
        for (int kk = 0; kk < KPAD / 4; ++kk) {
            bf[ct][kk].x = Bw[(size_t)(4 * kk + kb) * C0v + ct * 16 + mn];
            bf[ct][kk].y = Bw[(size_t)(4 * kk + kb + 1) * C0v + ct * 16 + mn];
        }

    for (int rt = 0; rt < RT; ++rt) {
        const int tile = base + rt;
        if (tile >= ntiles) break;
        const float* Ap = As + rt * (16 * KPAD);
        v8f acc[4];
#pragma unroll
        for (int ct = 0; ct < 4; ++ct) acc[ct] = (v8f){};
#pragma unroll
        for (int kk = 0; kk < KPAD / 4; ++kk) {
            v2f a;
            a.x = Ap[mn * KPAD + 4 * kk + kb];
            a.y = Ap[mn * KPAD + 4 * kk + kb + 1];
#pragma unroll
            for (int ct = 0; ct < 4; ++ct)
                acc[ct] = __builtin_amdgcn_wmma_f32_16x16x4_f32(false, a, false, bf[ct][kk],
                                                                (short)0, acc[ct], false, false);
        }
#pragma unroll
        for (int ct = 0; ct < 4; ++ct)
#pragma unroll
            for (int g = 0; g < 8; ++g) {
                int m = g + 8 * hl;
                y0[(size_t)(tile * 16 + m) * C0v + ct * 16 + mn] = acc[ct][g];
            }
    }
}

// ---------------- GEMM 1: [bnrelu(y0) | vmax[v]] (N x 128) @ W1(128 x 128) -> y1 ----------------
// 2 waves/block, 4 row-tiles/wave (64 rows in LDS); B register-blocked per col-tile;
// fast path unrolls all RT accumulators inside the K loop for 4-way WMMA ILP.
__global__ __launch_bounds__(64) void k_gemm1(const float* __restrict__ y0,
                                              const float* __restrict__ vmax,
                                              const int* __restrict__ coors,
                                              const float* __restrict__ scale0,
                                              const float* __restrict__ shift0,
                                              const float* __restrict__ W1,
                                              float* __restrict__ y1, int ntiles) {
    extern __shared__ float dynsmem[];                       // GW * RT * 16 * 128 floats
    const int lane = threadIdx.x & 31, wv = threadIdx.x >> 5;
    const int base = (blockIdx.x * GW + wv) * RT;
    float* As = dynsmem + wv * (RT * 16 * C1v);
    const int hl = lane >> 4, mn = lane & 15, kb = hl * 2;

    // stage: lane owns one float4 column group (K/4 == 32 == wave width)
    {
        const int c = lane * 4;
#pragma unroll
        for (int rt = 0; rt < RT; ++rt) {
            if (base + rt >= ntiles) break;
            if (c < C0v) {
                float s0 = scale0[c], s1 = scale0[c + 1], s2 = scale0[c + 2], s3 = scale0[c + 3];
                float h0 = shift0[c], h1 = shift0[c + 1], h2 = shift0[c + 2], h3 = shift0[c + 3];
                for (int row = 0; row < 16; ++row) {
                    size_t r = (size_t)(base + rt) * 16 + row;
                    float4 yv = ((const float4*)y0)[(r * C0v + c) >> 2];
                    float4 o;
                    o.x = fmaxf(yv.x * s0 + h0, 0.0f);
                    o.y = fmaxf(yv.y * s1 + h1, 0.0f);
                    o.z = fmaxf(yv.z * s2 + h2, 0.0f);
                    o.w = fmaxf(yv.w * s3 + h3, 0.0f);
                    ((float4*)As)[(rt * 16 * C1v + row * C1v + c) >> 2] = o;
                }
            } else {
                const int cc = c - C0v;
                for (int row = 0; row < 16; ++row) {
                    int r = (base + rt) * 16 + row;
                    int v = voxel_of(coors, r);
                    ((float4*)As)[(rt * 16 * C1v + row * C1v + c) >> 2] =
                        ((const float4*)vmax)[((size_t)v * C0v + cc) >> 2];
                }
            }
        }
    }
    __syncthreads();
    if (base >= ntiles) return;

    const bool full = (base + RT) <= ntiles;                 // wave-uniform
    for (int ct = 0; ct < C1v / 16; ++ct) {
        // hoist this col-tile's B fragments (reused across RT row tiles)
        v2f bf[C1v / 4];
#pragma unroll
        for (int kk = 0; kk < C1v / 4; ++kk) {
            bf[kk].x = W1[(size_t)(4 * kk + kb) * C1v + ct * 16 + mn];
            bf[kk].y = W1[(size_t)(4 * kk + kb + 1) * C1v + ct * 16 + mn];
        }
        if (full) {
            // fast path: RT independent accumulation chains, shared B fragments
            v8f acc[RT];
#pragma unroll
            for (int rt = 0; rt < RT; ++rt) acc[rt] = (v8f){};
#pragma unroll
            for (int kk = 0; kk < C1v / 4; ++kk) {
#pragma unroll
                for (int rt = 0; rt < RT; ++rt) {
                    v2f a;
                    a.x = As[rt * 16 * C1v + mn * C1v + 4 * kk + kb];
                    a.y = As[rt * 16 * C1v + mn * C1v + 4 * kk + kb + 1];
                    acc[rt] = __builtin_amdgcn_wmma_f32_16x16x4_f32(false, a, false, bf[kk],
                                                                    (short)0, acc[rt], false, false);
                }
            }
#pragma unroll
            for (int rt = 0; rt < RT; ++rt)
#pragma unroll
                for (int g = 0; g < 8; ++g) {
                    int m = g + 8 * hl;
                    y1[(size_t)((base + rt) * 16 + m) * C1v + ct * 16 + mn] = acc[rt][g];
                }
        } else {
            // tail path (last wave-group only)
            for (int rt = 0; rt < RT; ++rt) {
                const int tile = base + rt;
                if (tile >= ntiles) break;
                const float* Ap = As + rt * (16 * C1v);
                v8f acc = {};
#pragma unroll
                for (int kk = 0; kk < C1v / 4; ++kk) {
                    v2f a;
                    a.x = Ap[mn * C1v + 4 * kk + kb];
                    a.y = Ap[mn * C1v + 4 * kk + kb + 1];
                    acc = __builtin_amdgcn_wmma_f32_16x16x4_f32(false, a, false, bf[kk],
                                                                (short)0, acc, false, false);
                }
#pragma unroll
                for (int g = 0; g < 8; ++g) {
                    int m = g + 8 * hl;
                    y1[(size_t)(tile * 16 + m) * C1v + ct * 16 + mn] = acc[g];
                }
            }
        }
    }
}

// ---------------- GEMM 2: bnrelu(y1) @ Ws[0:128] + f_spatial @ Ws[128:130] + bs, ReLU -> out ----------------
__global__ __launch_bounds__(64) void k_gemm2(const float* __restrict__ y1,
                                              const float* __restrict__ scale1,
                                              const float* __restrict__ shift1,
                                              const float* __restrict__ Ws,
                                              const float* __restrict__ bs,
                                              const float* __restrict__ fsp,
                                              float* __restrict__ out, int ntiles) {
    extern __shared__ float dynsmem[];
    const int lane = threadIdx.x & 31, wv = threadIdx.x >> 5;
    const int base = (blockIdx.x * GW + wv) * RT;
    float* As = dynsmem + wv * (RT * 16 * C1v);
    const int hl = lane >> 4, mn = lane & 15, kb = hl * 2;

    {
        const int c = lane * 4;
        float s0 = scale1[c], s1 = scale1[c + 1], s2 = scale1[c + 2], s3 = scale1[c + 3];
        float h0 = shift1[c], h1 = shift1[c + 1], h2 = shift1[c + 2], h3 = shift1[c + 3];
#pragma unroll
        for (int rt = 0; rt < RT; ++rt) {
            if (base + rt >= ntiles) break;
            for (int row = 0; row < 16; ++row) {
                size_t r = (size_t)(base + rt) * 16 + row;
                float4 yv = ((const float4*)y1)[(r * C1v + c) >> 2];
                float4 o;
                o.x = fmaxf(yv.x * s0 + h0, 0.0f);
                o.y = fmaxf(yv.y * s1 + h1, 0.0f);
                o.z = fmaxf(yv.z * s2 + h2, 0.0f);
                o.w = fmaxf(yv.w * s3 + h3, 0.0f);
                ((float4*)As)[(rt * 16 * C1v + row * C1v + c) >> 2] = o;
            }
        }
    }
    __syncthreads();
    if (base >= ntiles) return;

    const bool full = (base + RT) <= ntiles;                 // wave-uniform
    for (int ct = 0; ct < C1v / 16; ++ct) {
        v2f bf[C1v / 4];
#pragma unroll
        for (int kk = 0; kk < C1v / 4; ++kk) {
            bf[kk].x = Ws[(size_t)(4 * kk + kb) * C1v + ct * 16 + mn];
            bf[kk].y = Ws[(size_t)(4 * kk + kb + 1) * C1v + ct * 16 + mn];
        }
        const int col = ct * 16 + mn;
        const float w128 = Ws[(size_t)128 * C1v + col];
        const float w129 = Ws[(size_t)129 * C1v + col];
        const float bb = bs[col];
        if (full) {
            v8f acc[RT];
#pragma unroll
            for (int rt = 0; rt < RT; ++rt) acc[rt] = (v8f){};
#pragma unroll
            for (int kk = 0; kk < C1v / 4; ++kk) {
#pragma unroll
                for (int rt = 0; rt < RT; ++rt) {
                    v2f a;
                    a.x = As[rt * 16 * C1v + mn * C1v + 4 * kk + kb];
                    a.y = As[rt * 16 * C1v + mn * C1v + 4 * kk + kb + 1];
                    acc[rt] = __builtin_amdgcn_wmma_f32_16x16x4_f32(false, a, false, bf[kk],
                                                                    (short)0, acc[rt], false, false);
                }
            }
#pragma unroll
            for (int rt = 0; rt < RT; ++rt)
#pragma unroll
                for (int g = 0; g < 8; ++g) {
                    int m = g + 8 * hl;
                    size_t r = (size_t)(base + rt) * 16 + m;
                    float2 fs = ((const float2*)fsp)[r];
                    float val = acc[rt][g] + fs.x * w128 + fs.y * w129 + bb;
                    out[r * C1v + col] = fmaxf(val, 0.0f);
                }
        } else {
            for (int rt = 0; rt < RT; ++rt) {
                const int tile = base + rt;
                if (tile >= ntiles) break;
                const float* Ap = As + rt * (16 * C1v);
                v8f acc = {};
#pragma unroll
                for (int kk = 0; kk < C1v / 4; ++kk) {
                    v2f a;
                    a.x = Ap[mn * C1v + 4 * kk + kb];
                    a.y = Ap[mn * C1v + 4 * kk + kb + 1];
                    acc = __builtin_amdgcn_wmma_f32_16x16x4_f32(false, a, false, bf[kk],
                                                                (short)0, acc, false, false);
                }
#pragma unroll
                for (int g = 0; g < 8; ++g) {
                    int m = g + 8 * hl;
                    size_t r = (size_t)tile * 16 + m;
                    float2 fs = ((const float2*)fsp)[r];
                    float val = acc[g] + fs.x * w128 + fs.y * w129 + bb;
                    out[r * C1v + col] = fmaxf(val, 0.0f);
                }
            }
        }
    }
}

// ---------------- host launcher ----------------
extern "C" void kernel_launch(void* const* d_in, const int* in_sizes, int n_in,
                              void* d_out, int out_size, void* d_ws, size_t ws_size,
                              hipStream_t stream) {
    const float* feat = (const float*)d_in[0];
    const int*   coors = (const int*)d_in[1];
    const float* W0 = (const float*)d_in[2];
    const float* g0 = (const float*)d_in[3];
    const float* b0 = (const float*)d_in[4];
    const float* W1 = (const float*)d_in[5];
    const float* g1 = (const float*)d_in[6];
    const float* b1 = (const float*)d_in[7];
    const float* Ws = (const float*)d_in[8];
    const float* bs = (const float*)d_in[9];
    float* out = (float*)d_out;
    float* w = (float*)d_ws;

    // workspace layout (float offsets)
    const size_t o_cnt = 0;
    const size_t o_sx  = (size_t)CANVAS;
    const size_t o_sy  = 2 * (size_t)CANVAS;
    const size_t o_sz  = 3 * (size_t)CANVAS;
    const size_t o_pd  = 4 * (size_t)CANVAS;
    const size_t o_vmx = 5 * (size_t)CANVAS;                 // CANVAS * 64
    const size_t o_bn  = o_vmx + 64 * (size_t)CANVAS;
    const size_t o_sum0 = o_bn,        o_sq0 = o_bn + 64;
    const size_t o_sum1 = o_bn + 128,  o_sq1 = o_bn + 256;
    const size_t o_sc0 = o_bn + 384,   o_sh0 = o_bn + 448;
    const size_t o_sc1 = o_bn + 512,   o_sh1 = o_bn + 640;
    const size_t o_w0p = o_bn + 768;                          // 16*64
    const size_t o_xp  = o_w0p + KPAD * C0v;                  // N*16
    const size_t o_fs  = o_xp + (size_t)NPTS * KPAD;          // N*2
    const size_t o_y0  = o_fs + (size_t)NPTS * 2;             // N*64
    const size_t o_y1  = o_y0 + (size_t)NPTS * C0v;           // N*128

    const size_t nzero = o_bn + 384;   // counts..pdist + vmax + BN accumulators
    const int TB = 256;
    const int nPtBlk = (NPTS + TB - 1) / TB;
    const int bigBlocks = (NWT + GW - 1) / GW;                // 3907
    const size_t bigSmem = (size_t)GW * RT * 16 * C1v * sizeof(float);  // 64 KB dynamic

    k_zero<<<4096, TB, 0, stream>>>(w, nzero);
    k_pad_w0<<<(KPAD * C0v + TB - 1) / TB, TB, 0, stream>>>(W0, w + o_w0p);
    k_scatter_stats<<<nPtBlk, TB, 0, stream>>>(feat, coors, w + o_cnt, w + o_sx, w + o_sy, w + o_sz, NPTS);
    k_scatter_pdist<<<nPtBlk, TB, 0, stream>>>(feat, coors, w + o_cnt, w + o_sx, w + o_sy, w + o_sz, w + o_pd, NPTS);
    k_assemble<<<nPtBlk, TB, 0, stream>>>(feat, coors, w + o_cnt, w + o_sx, w + o_sy, w + o_sz, w + o_pd,
                                          w + o_xp, w + o_fs, NPTS);
    k_gemm0<<<(NTILES + 4 * RT - 1) / (4 * RT), 128, 0, stream>>>(w + o_xp, w + o_w0p, w + o_y0, NTILES);
    k_bnstats<<<256, 256, 0, stream>>>(w + o_y0, w + o_sum0, w + o_sq0, C0v, NPTS);
    k_bnfinal<<<1, C0v, 0, stream>>>(w + o_sum0, w + o_sq0, g0, b0, w + o_sc0, w + o_sh0, C0v, 1.0f / NPTS);
    k_vmax<<<(NPTS * C0v + TB - 1) / TB, TB, 0, stream>>>(w + o_y0, w + o_sc0, w + o_sh0, coors, w + o_vmx, NPTS);
    k_gemm1<<<bigBlocks, GW * 32, bigSmem, stream>>>(w + o_y0, w + o_vmx, coors, w + o_sc0, w + o_sh0, W1,
                                                     w + o_y1, NTILES);
    k_bnstats<<<256, 256, 0, stream>>>(w + o_y1, w + o_sum1, w + o_sq1, C1v, NPTS);
    k_bnfinal<<<1, C1v, 0, stream>>>(w + o_sum1, w + o_sq1, g1, b1, w + o_sc1, w + o_sh1, C1v, 1.0f / NPTS);
    k_gemm2<<<bigBlocks, GW * 32, bigSmem, stream>>>(w + o_y1, w + o_sc1, w + o_sh1, Ws, bs, w + o_fs, out, NTILES);
}